// Generator_causal_65962107732642
// MI455X (gfx1250) — compile-verified
//
#include <hip/hip_runtime.h>
#include <hip/hip_bf16.h>

// CDNA5 / gfx1250: wave32, WMMA f32 16x16x4 (exact fp32 matrix math).
typedef float v2f __attribute__((ext_vector_type(2)));
typedef float v8f __attribute__((ext_vector_type(8)));

#define XD    128          // x_dim (nodes)
#define HD    128          // h_dim
#define PITCH 132          // LDS row pitch in floats (conflict-free for the frag pattern)
#define BT    32           // batch rows per workgroup
#define NTH   256          // 8 wave32

// WMMA fragment layout assumptions (per CDNA5 ISA):
//  A 16x4 f32 : lane L<16  -> (M=L,    K=0..1), lane>=16 -> (M=L-16, K=2..3)  [2 VGPR = v2f]
//  B 4x16 f32 : lane L<16  -> (K=0..1, N=L),    lane>=16 -> (K=2..3, N=L-16)  [mirror of A]
//  C/D 16x16  : VGPR v, lane L -> (M = v + 8*(L>=16), N = L%16)

__global__ __launch_bounds__(NTH, 1)
void causal_dag_kernel(const float* __restrict__ x,  const float* __restrict__ z,
                       const float* __restrict__ M,  const float* __restrict__ Wi,
                       const float* __restrict__ bi, const float* __restrict__ Wf,
                       const float* __restrict__ bf, const float* __restrict__ W1,
                       const float* __restrict__ b1, const float* __restrict__ W2,
                       const float* __restrict__ b2, float* __restrict__ out)
{
    __shared__ float s_out[BT * PITCH];   // resident out tile (updated column-by-column)
    __shared__ float s_h1 [BT * PITCH];   // h after fc_i / after layer2
    __shared__ float s_h2 [BT * PITCH];   // xm / h after layer1
    __shared__ float s_mask[XD];
    __shared__ float s_wf [HD];
    __shared__ float s_part[BT * 8];
    __shared__ float s_bf;

    const int    t    = threadIdx.x;
    const int    lane = t & 31;
    const int    wv   = t >> 5;            // wave id 0..7 -> N-tile
    const int    lr   = lane & 15;
    const int    half = lane >> 4;         // 0 or 1
    const int    koff = half * 2;          // K offset within a K=4 chunk
    const int    n    = wv * 16 + lr;      // this lane's hidden/output column
    const size_t Bbase = (size_t)blockIdx.x * BT;

    // ---- load x tile into resident out tile ----
    for (int q = t; q < BT * XD; q += NTH) {
        int m = q >> 7, c = q & 127;
        s_out[m * PITCH + c] = x[(Bbase + m) * (size_t)XD + c];
    }

    // ---- hoist shared-MLP B fragments into registers (reused 128x) ----
    v2f w1f[32], w2f[32];
    {
        const float* r1 = W1 + (size_t)n * HD;   // B[k][n] = W1[n][k]
        const float* r2 = W2 + (size_t)n * HD;
#pragma unroll
        for (int kk = 0; kk < 32; ++kk) {
            w1f[kk] = *(const v2f*)(r1 + 4 * kk + koff);   // 8B aligned (even offsets)
            w2f[kk] = *(const v2f*)(r2 + 4 * kk + koff);
        }
    }
    const float b1n = b1[n];
    const float b2n = b2[n];

    __syncthreads();

#pragma unroll 1
    for (int i = 0; i < XD; ++i) {
        // ---- stage 0: per-node vectors ----
        if (t < XD) {
            s_mask[t] = (t == i) ? 0.0f : M[(size_t)t * XD + i];   // M[:, i], self zeroed
        } else {
            s_wf[t - XD] = Wf[(size_t)i * HD + (t - XD)];
        }
        if (t == 0) s_bf = bf[i];
        __syncthreads();

        // ---- stage 1: xm = out * mask ----
        for (int q = t; q < BT * XD; q += NTH) {
            int m = q >> 7, c = q & 127;
            s_h2[m * PITCH + c] = s_out[m * PITCH + c] * s_mask[c];
        }
        __syncthreads();

        // ---- stage 2: GEMM1  h1 = relu(xm @ WiX_i^T + z_i (x) wz_i + bi_i) ----
        const float* wiRow = Wi + ((size_t)i * HD + n) * (XD + 1);   // Wi[i][n][:]
        const float  wzn   = wiRow[XD];
        const float  bin   = bi[(size_t)i * HD + n];
        // Prefetch next node's weight row (global_prefetch_b8 on gfx1250)
        const int inext = (i + 1 < XD) ? (i + 1) : i;
        __builtin_prefetch(Wi + ((size_t)inext * HD + n) * (XD + 1), 0, 1);

        v8f acc0, acc1;
#pragma unroll
        for (int v = 0; v < 8; ++v) { acc0[v] = 0.0f; acc1[v] = 0.0f; }

#pragma unroll
        for (int kk = 0; kk < 32; ++kk) {
            v2f b;                                     // B[k][n] = Wi[i][n][k]; row pitch 129
            b.x = wiRow[4 * kk + koff];                // scalar loads: pitch is odd -> no b64
            b.y = wiRow[4 * kk + koff + 1];
            v2f a0 = *(const v2f*)(s_h2 + (0  + lr) * PITCH + 4 * kk + koff);
            v2f a1 = *(const v2f*)(s_h2 + (16 + lr) * PITCH + 4 * kk + koff);
            acc0 = __builtin_amdgcn_wmma_f32_16x16x4_f32(false, a0, false, b, (short)0, acc0, false, false);
            acc1 = __builtin_amdgcn_wmma_f32_16x16x4_f32(false, a1, false, b, (short)0, acc1, false, false);
        }
#pragma unroll
        for (int v = 0; v < 8; ++v) {
            int   m0 = v + 8 * half;
            int   m1 = 16 + m0;
            float z0 = z[(Bbase + m0) * (size_t)XD + i];
            float z1 = z[(Bbase + m1) * (size_t)XD + i];
            s_h1[m0 * PITCH + n] = fmaxf(acc0[v] + z0 * wzn + bin, 0.0f);
            s_h1[m1 * PITCH + n] = fmaxf(acc1[v] + z1 * wzn + bin, 0.0f);
        }
        __syncthreads();

        // ---- stage 3: GEMM2  h2 = relu(h1 @ W1^T + b1) ----
#pragma unroll
        for (int v = 0; v < 8; ++v) { acc0[v] = 0.0f; acc1[v] = 0.0f; }
#pragma unroll
        for (int kk = 0; kk < 32; ++kk) {
            v2f a0 = *(const v2f*)(s_h1 + (0  + lr) * PITCH + 4 * kk + koff);
            v2f a1 = *(const v2f*)(s_h1 + (16 + lr) * PITCH + 4 * kk + koff);
            acc0 = __builtin_amdgcn_wmma_f32_16x16x4_f32(false, a0, false, w1f[kk], (short)0, acc0, false, false);
            acc1 = __builtin_amdgcn_wmma_f32_16x16x4_f32(false, a1, false, w1f[kk], (short)0, acc1, false, false);
        }
#pragma unroll
        for (int v = 0; v < 8; ++v) {
            int m0 = v + 8 * half;
            int m1 = 16 + m0;
            s_h2[m0 * PITCH + n] = fmaxf(acc0[v] + b1n, 0.0f);
            s_h2[m1 * PITCH + n] = fmaxf(acc1[v] + b1n, 0.0f);
        }
        __syncthreads();

        // ---- stage 4: GEMM3  h1 = relu(h2 @ W2^T + b2) ----
#pragma unroll
        for (int v = 0; v < 8; ++v) { acc0[v] = 0.0f; acc1[v] = 0.0f; }
#pragma unroll
        for (int kk = 0; kk < 32; ++kk) {
            v2f a0 = *(const v2f*)(s_h2 + (0  + lr) * PITCH + 4 * kk + koff);
            v2f a1 = *(const v2f*)(s_h2 + (16 + lr) * PITCH + 4 * kk + koff);
            acc0 = __builtin_amdgcn_wmma_f32_16x16x4_f32(false, a0, false, w2f[kk], (short)0, acc0, false, false);
            acc1 = __builtin_amdgcn_wmma_f32_16x16x4_f32(false, a1, false, w2f[kk], (short)0, acc1, false, false);
        }
#pragma unroll
        for (int v = 0; v < 8; ++v) {
            int m0 = v + 8 * half;
            int m1 = 16 + m0;
            s_h1[m0 * PITCH + n] = fmaxf(acc0[v] + b2n, 0.0f);
            s_h1[m1 * PITCH + n] = fmaxf(acc1[v] + b2n, 0.0f);
        }
        __syncthreads();

        // ---- stage 5: fc_f  o[m] = sum_h h1[m][h]*Wf[i][h] + bf[i] ----
        {
            int m = t >> 3, seg = t & 7;
            const float* hrow = s_h1 + m * PITCH + seg * 16;
            const float* wfp  = s_wf + seg * 16;
            float p = 0.0f;
#pragma unroll
            for (int j = 0; j < 16; ++j) p += hrow[j] * wfp[j];
            s_part[t] = p;                      // t == m*8 + seg
        }
        __syncthreads();
        if (t < BT) {
            float o = s_bf;
#pragma unroll
            for (int s = 0; s < 8; ++s) o += s_part[t * 8 + s];
            s_out[t * PITCH + i] = o;           // feed back into resident out tile
        }
        __syncthreads();
    }

    // ---- final store of the out tile ----
    for (int q = t; q < BT * XD; q += NTH) {
        int m = q >> 7, c = q & 127;
        out[(Bbase + m) * (size_t)XD + c] = s_out[m * PITCH + c];
    }
}

extern "C" void kernel_launch(void* const* d_in, const int* in_sizes, int n_in,
                              void* d_out, int out_size, void* d_ws, size_t ws_size,
                              hipStream_t stream) {
    const float* x  = (const float*)d_in[0];
    const float* z  = (const float*)d_in[1];
    const float* M  = (const float*)d_in[2];
    const float* Wi = (const float*)d_in[3];
    const float* bi = (const float*)d_in[4];
    const float* Wf = (const float*)d_in[5];
    const float* bf = (const float*)d_in[6];
    const float* W1 = (const float*)d_in[7];
    const float* b1 = (const float*)d_in[8];
    const float* W2 = (const float*)d_in[9];
    const float* b2 = (const float*)d_in[10];
    float* outp = (float*)d_out;

    const int Btotal = in_sizes[0] / XD;       // 131072
    const int blocks = Btotal / BT;            // 4096 workgroups
    causal_dag_kernel<<<blocks, NTH, 0, stream>>>(x, z, M, Wi, bi, Wf, bf, W1, b1, W2, b2, outp);
    (void)d_ws; (void)ws_size; (void)n_in; (void)out_size;
}